// VecNodesConv_58523224375827
// MI455X (gfx1250) — compile-verified
//
#include <hip/hip_runtime.h>

// Problem constants (from reference)
#define N_NODES      50000
#define N_EDGES      1600000
#define DIM          64
#define COLS         (N_NODES * 3)     // 150000 (node, d) columns
#define COL_TILES    (COLS / 16)       // 9375 exact
#define NODE_STRIDE  (DIM * 3)         // 192 floats per node
#define INV_SQRT_2F  0.70710678118654752440f

typedef __attribute__((ext_vector_type(2))) float v2f;
typedef __attribute__((ext_vector_type(8))) float v8f;

// ---------------------------------------------------------------------------
// Kernel 1: per-node dual matmul Y = W * X (W: 64x64, X: 64x3) using
// V_WMMA_F32_16X16X4_F32. One wave per 16-column tile (columns = node*3+d).
//
// A (16x4 f32) per-lane layout: lanes 0-15 hold M=lane, v0=K0,v1=K1;
// lanes 16-31 hold M=lane-16, v0=K2,v1=K3.  -> contiguous v2f load from W row.
// B (4x16 f32) mirrored: v0/v1 hold rows (K) = half*2, half*2+1 at N=lane&15.
// C/D (16x16 f32): VGPR r -> M=r (+8 for upper half-wave), N = lane&15.
// ---------------------------------------------------------------------------
__global__ __launch_bounds__(256) void vnc_transform_wmma(
    const float* __restrict__ x,       // [N_NODES][64][3]
    const float* __restrict__ Wn,      // [64][64]
    const float* __restrict__ We,      // [64][64]
    float* __restrict__ y_node_out,    // d_out: y_node * 1/sqrt(2)
    float* __restrict__ y_edge_out)    // ws:    y_edge (unscaled)
{
    const int wave = (int)((blockIdx.x * blockDim.x + threadIdx.x) >> 5);
    if (wave >= COL_TILES) return;     // wave-uniform exit (EXEC stays all-1 otherwise)

    const int lane    = (int)(threadIdx.x & 31);
    const int halfSel = lane >> 4;            // 0: K pair {0,1}; 1: K pair {2,3}
    const int col     = wave * 16 + (lane & 15);
    const int node    = col / 3;
    const int dcomp   = col - node * 3;
    const float* xcol = x + node * NODE_STRIDE + dcomp;   // X[i] at xcol[i*3]

    const v8f z = {0.f, 0.f, 0.f, 0.f, 0.f, 0.f, 0.f, 0.f};
    v8f accN[4], accE[4];
#pragma unroll
    for (int m = 0; m < 4; ++m) { accN[m] = z; accE[m] = z; }

    // K = 64 in 16 steps of K=4
#pragma unroll 4
    for (int k = 0; k < 16; ++k) {
        const int i0 = (k << 2) + (halfSel << 1);         // first of this lane's K pair
        v2f b;
        b.x = xcol[i0 * 3];
        b.y = xcol[(i0 + 1) * 3];

        const int acol = (k << 2) + (halfSel << 1);       // even -> 8B aligned
        const float* wnp = Wn + (lane & 15) * DIM + acol;
        const float* wep = We + (lane & 15) * DIM + acol;
#pragma unroll
        for (int m = 0; m < 4; ++m) {
            const v2f an = *(const v2f*)(wnp + m * 16 * DIM);
            const v2f ae = *(const v2f*)(wep + m * 16 * DIM);
            accN[m] = __builtin_amdgcn_wmma_f32_16x16x4_f32(
                false, an, false, b, (short)0, accN[m], false, false);
            accE[m] = __builtin_amdgcn_wmma_f32_16x16x4_f32(
                false, ae, false, b, (short)0, accE[m], false, false);
        }
    }

    // Store D tiles: element r of accumulator = output row o = m*16 + half*8 + r
    float* oc = y_node_out + node * NODE_STRIDE + dcomp;
    float* ec = y_edge_out + node * NODE_STRIDE + dcomp;
#pragma unroll
    for (int m = 0; m < 4; ++m) {
#pragma unroll
        for (int r = 0; r < 8; ++r) {
            const int o = m * 16 + halfSel * 8 + r;
            oc[o * 3] = accN[m][r] * INV_SQRT_2F;
            ec[o * 3] = accE[m][r];
        }
    }
}

// ---------------------------------------------------------------------------
// Kernel 2: edge gather + scatter-add.  out[dst] += y_edge[src] * nc[dst]/sqrt2
// One thread per (edge, float4 chunk): coalesced 16B gather (L2-resident:
// y_edge = 38.4MB << 192MB L2), then 4 non-returning relaxed f32 atomics
// (resolve at L2 atomic units).
// ---------------------------------------------------------------------------
__global__ __launch_bounds__(256) void vnc_scatter_edges(
    const float* __restrict__ y_edge,
    const int*   __restrict__ src,
    const int*   __restrict__ dst,
    const float* __restrict__ norm_coeff,
    float* __restrict__ out)
{
    const int t = (int)(blockIdx.x * blockDim.x + threadIdx.x);
    const int e = t / 48;                 // 48 float4 chunks per edge (192 floats)
    if (e >= N_EDGES) return;
    const int q = t - e * 48;

    const int s  = src[e];
    const int dd = dst[e];
    const float scale = norm_coeff[dd] * INV_SQRT_2F;

    const float4 v = *((const float4*)(y_edge + s * NODE_STRIDE) + q);
    float* p = out + dd * NODE_STRIDE + q * 4;

    __hip_atomic_fetch_add(p + 0, v.x * scale, __ATOMIC_RELAXED, __HIP_MEMORY_SCOPE_AGENT);
    __hip_atomic_fetch_add(p + 1, v.y * scale, __ATOMIC_RELAXED, __HIP_MEMORY_SCOPE_AGENT);
    __hip_atomic_fetch_add(p + 2, v.z * scale, __ATOMIC_RELAXED, __HIP_MEMORY_SCOPE_AGENT);
    __hip_atomic_fetch_add(p + 3, v.w * scale, __ATOMIC_RELAXED, __HIP_MEMORY_SCOPE_AGENT);
}

extern "C" void kernel_launch(void* const* d_in, const int* in_sizes, int n_in,
                              void* d_out, int out_size, void* d_ws, size_t ws_size,
                              hipStream_t stream) {
    (void)in_sizes; (void)n_in; (void)out_size; (void)ws_size;

    const float* x   = (const float*)d_in[0];   // (1, 50000, 64, 3)
    const float* Wn  = (const float*)d_in[1];   // (64, 64)
    const float* We  = (const float*)d_in[2];   // (64, 64)
    const float* nc  = (const float*)d_in[3];   // (50000,)
    const int*   src = (const int*)  d_in[4];   // (1.6M,)
    const int*   dst = (const int*)  d_in[5];   // (1.6M,)

    float* out    = (float*)d_out;              // (1, 50000, 64, 3)
    float* y_edge = (float*)d_ws;               // 38.4 MB scratch

    // Pass 1: WMMA transforms. 9375 column tiles, 8 waves / 256-thread block.
    const int blocks1 = (COL_TILES + 7) / 8;    // 1172
    vnc_transform_wmma<<<blocks1, 256, 0, stream>>>(x, Wn, We, out, y_edge);

    // Pass 2: gather + atomic scatter. 1.6M edges * 48 chunks = 76.8M threads.
    const long long total = (long long)N_EDGES * 48;
    const int blocks2 = (int)((total + 255) / 256);  // 300000
    vnc_scatter_edges<<<blocks2, 256, 0, stream>>>(y_edge, src, dst, nc, out);
}